// HashEncoding_64390149701955
// MI455X (gfx1250) — compile-verified
//
#include <hip/hip_runtime.h>

typedef float v2f __attribute__((ext_vector_type(2)));
typedef float v4f __attribute__((ext_vector_type(4)));
typedef v4f   v4f_a8 __attribute__((aligned(8)));   // 16B load at 8B alignment (pair gather)
typedef int   v2i __attribute__((__vector_size__(2 * sizeof(int))));

#if __has_builtin(__builtin_amdgcn_global_load_async_to_lds_b64)
#define USE_ASYNC_LDS 1
#else
#define USE_ASYNC_LDS 0
#endif

// Level-0 dense table: grid coords <= 16, max flat index 16 + 16*16 + 16*256 = 4368.
// Stage 4608 = 18*256 entries so the copy loop has a uniform, branch-free trip count
// (reads stay inside level 0's 2^19-entry table).
#define NL0_STAGE_ITERS 18
#define NL0 (NL0_STAGE_ITERS * 256)   // 4608 entries, 36,864 B LDS

__global__ __launch_bounds__(256) void hashgrid_encode_kernel(
    const float* __restrict__ coords,   // [N,3]
    const float* __restrict__ table,    // [16, 2^19, 2]
    float* __restrict__ out,            // [N,32]
    int nPoints)
{
    constexpr int      L    = 16;
    constexpr unsigned T    = 1u << 19;
    constexpr unsigned MASK = T - 1u;
    constexpr unsigned P1   = 2654435761u;  // tcnn prime dim1
    constexpr unsigned P2   = 805459861u;   // tcnn prime dim2
    // scale_l = 16 * (2^(7/15))^l - 1
    constexpr float kScale[16] = {
        15.0f,        21.1106054f,  29.5549296f,  41.2242515f,
        57.3502355f,  79.6349472f,  110.430469f,  152.987196f,
        211.796900f,  293.066779f,  405.374669f,  560.574374f,
        775.046882f,  1071.42915f,  1481.00339f,  2047.0f };
    constexpr unsigned kRes[5] = {16u, 23u, 31u, 43u, 59u};  // dense levels (res^3 <= T)

    __shared__ v2f sL0[NL0];  // level-0 dense table slice

    // ---- Stage level-0 table into LDS (gfx1250 async DMA path, ASYNCcnt) ----
    {
        const v2f* __restrict__ g0 = (const v2f*)table;
#pragma unroll
        for (int i = 0; i < NL0_STAGE_ITERS; ++i) {
            const int e = (int)threadIdx.x + i * 256;
#if USE_ASYNC_LDS
            __builtin_amdgcn_global_load_async_to_lds_b64(
                (__attribute__((address_space(1))) v2i*)((const v2i*)(g0 + e)),
                (__attribute__((address_space(3))) v2i*)((v2i*)(&sL0[e])),
                0, 0);
#else
            sL0[e] = g0[e];
#endif
        }
#if USE_ASYNC_LDS
#if __has_builtin(__builtin_amdgcn_s_wait_asynccnt)
        __builtin_amdgcn_s_wait_asynccnt(0);
#else
        asm volatile("s_wait_asynccnt 0" ::: "memory");
#endif
#endif
    }
    __syncthreads();

    const int n = (int)(blockIdx.x * 256u + threadIdx.x);
    if (n >= nPoints) return;

    // Non-temporal coord loads: read-once stream, don't pollute L2 (table lives there).
    const float* cp = coords + (size_t)n * 3;
    const float cx = (__builtin_nontemporal_load(cp + 0) + 1.0f) * 0.5f;
    const float cy = (__builtin_nontemporal_load(cp + 1) + 1.0f) * 0.5f;
    const float cz = (__builtin_nontemporal_load(cp + 2) + 1.0f) * 0.5f;

    float acc[2 * L];

#pragma unroll
    for (int l = 0; l < L; ++l) {
        const float sc = kScale[l];
        const float px = fmaf(cx, sc, 0.5f);
        const float py = fmaf(cy, sc, 0.5f);
        const float pz = fmaf(cz, sc, 0.5f);
        const float fx = floorf(px), fy = floorf(py), fz = floorf(pz);
        const float tx = px - fx, ty = py - fy, tz = pz - fz;
        const unsigned gx = (unsigned)(int)fx;
        const unsigned gy = (unsigned)(int)fy;
        const unsigned gz = (unsigned)(int)fz;

        // Corner feature values; corner index c: bit2 = x-corner, bit1 = y, bit0 = z.
        v2f v[8];

        if (l == 0) {
            // Level 0 from LDS: 8 indexed gathers (compiler pairs x-corners into
            // ds_load_2addr_b64 since they are 8B apart).
            const unsigned r = 16u, r2 = 256u;
            const unsigned base = gx + gy * r + gz * r2;
#pragma unroll
            for (int c = 0; c < 8; ++c) {
                unsigned o = base;
                if ((c >> 2) & 1) o += 1u;
                if ((c >> 1) & 1) o += r;
                if (c & 1)        o += r2;
                v[c] = sL0[o];
            }
        } else if (l < 5) {
            // Dense level: x-corner pairs are adjacent entries -> one b128 load
            // fetches both corners (4 requests instead of 8).
            const unsigned r  = kRes[l];
            const unsigned r2 = r * r;
            const unsigned base = gx + gy * r + gz * r2;
            const v2f* __restrict__ tbl = (const v2f*)table + (size_t)l * T;
#pragma unroll
            for (int p = 0; p < 4; ++p) {       // p: bit1 = y-corner, bit0 = z-corner
                unsigned i0 = base;
                if ((p >> 1) & 1) i0 += r;
                if (p & 1)        i0 += r2;
                const v4f_a8 q = *(const v4f_a8*)(tbl + i0);
                v[p]     = {q.x, q.y};          // x = 0 corner
                v[p + 4] = {q.z, q.w};          // x = 1 corner
            }
        } else {
            // Hashed level: (x*1) ^ (y*P1) ^ (z*P2) & (T-1); 8 independent b64
            // gathers, all indices computed before the loads for MLP.
            const unsigned x0 = gx,      x1 = gx + 1u;
            const unsigned y0 = gy * P1, y1 = y0 + P1;
            const unsigned z0 = gz * P2, z1 = z0 + P2;
            unsigned idx[8];
#pragma unroll
            for (int c = 0; c < 8; ++c) {
                const unsigned h = ((((c >> 2) & 1) ? x1 : x0))
                                 ^ ((((c >> 1) & 1) ? y1 : y0))
                                 ^ (((c & 1)        ? z1 : z0));
                idx[c] = h & MASK;
            }
            const v2f* __restrict__ tbl = (const v2f*)table + (size_t)l * T;
#pragma unroll
            for (int c = 0; c < 8; ++c) v[c] = tbl[idx[c]];
        }

        // Trilinear blend
        const float wx0 = 1.0f - tx, wy0 = 1.0f - ty, wz0 = 1.0f - tz;
        float a0 = 0.0f, a1 = 0.0f;
#pragma unroll
        for (int c = 0; c < 8; ++c) {
            const float w = ((((c >> 2) & 1) ? tx : wx0))
                          * ((((c >> 1) & 1) ? ty : wy0))
                          * (((c & 1)        ? tz : wz0));
            a0 = fmaf(w, v[c].x, a0);
            a1 = fmaf(w, v[c].y, a1);
        }
        acc[2 * l + 0] = a0;
        acc[2 * l + 1] = a1;
    }

    // Full 128B row per thread via 8 non-temporal b128 stores; across a wave the
    // 8 stores tile a contiguous 4KB region (full-line writes, no L2 pollution).
    v4f* __restrict__ orow = (v4f*)(out + (size_t)n * 32);
#pragma unroll
    for (int i = 0; i < 8; ++i) {
        v4f t = { acc[4 * i + 0], acc[4 * i + 1], acc[4 * i + 2], acc[4 * i + 3] };
        __builtin_nontemporal_store(t, orow + i);
    }
}

extern "C" void kernel_launch(void* const* d_in, const int* in_sizes, int n_in,
                              void* d_out, int out_size, void* d_ws, size_t ws_size,
                              hipStream_t stream) {
    (void)n_in; (void)out_size; (void)d_ws; (void)ws_size;
    const float* coords = (const float*)d_in[0];
    const float* table  = (const float*)d_in[1];
    float* out = (float*)d_out;
    const int nPoints = in_sizes[0] / 3;   // [N,3] flat
    const int blocks = (nPoints + 255) / 256;
    hipLaunchKernelGGL(hashgrid_encode_kernel, dim3(blocks), dim3(256), 0, stream,
                       coords, table, out, nPoints);
}